// SHNMN_30331059044956
// MI455X (gfx1250) — compile-verified
//
#include <hip/hip_runtime.h>

// ---------------------------------------------------------------------------
// SHNMN forward for MI455X (gfx1250, wave32), fp32 WMMA path, fully fused:
// one kernel per module does  proj-GEMM -> LDS -> implicit-GEMM conv -> ReLU.
// All WMMA fragments are stored pair-interleaved in LDS so every A/B fragment
// is one aligned ds_load_b64 straight into an even VGPR pair (no repacking).
// ---------------------------------------------------------------------------

typedef __attribute__((ext_vector_type(2))) float v2f;
typedef __attribute__((ext_vector_type(8))) float v8f;

#define DIMC   64
#define NPIX   256                    // 16x16 pixels
#define CW     36864                  // 64*64*3*3 conv-weight floats
#define QDIM   45184
#define PWOFF  36928                  // proj_w offset in question_rep
#define PBOFF  45120                  // proj_b offset
#define NMOD   3
#define MSLOTS 5
#define HSM    16384                  // floats per (b, m) slot
#define HSB    81920                  // floats per sample in d_out
#define BATCH  128
#define PADP   324                    // 18*18 padded plane

// LDS partition (float offsets). Phase-2 weights overlay the dead x/pw region.
#define OFF_X    0                    // [64 pairs][256 pix][2]      = 32768
#define OFF_PW   32768                // [64 pairs][64 o][2]         =  8192
#define OFF_W    0                    // [32 pairs][9 t][64 o][2]    = 36864 (overlay)
#define OFF_INP  40960                // [32 pairs][18*18][2]        = 20736
#define OFF_PB   61696                // [64]
#define OFF_CB   61760                // [64]
#define LDS_FLOATS 61824              // ~241.5 KB  (<= 320 KB/WG on CDNA5)

__device__ __forceinline__ void softmaxn(const float* __restrict__ v, int n, float* w) {
  float mx = v[0];
  for (int k = 1; k < n; ++k) mx = fmaxf(mx, v[k]);
  float s = 0.f;
  for (int k = 0; k < n; ++k) { w[k] = __expf(v[k] - mx); s += w[k]; }
  float inv = 1.f / s;
  for (int k = 0; k < n; ++k) w[k] *= inv;
}

__device__ __forceinline__ v2f ldsp(const float* p) {     // aligned LDS pair load
  return *(const v2f*)p;
}

// h_prev init: slot 0 = zeros (sentinel), slot 1 = img.
__global__ void init_h_kernel(const float* __restrict__ img, float* __restrict__ hbuf) {
  int idx = blockIdx.x * blockDim.x + threadIdx.x;
  if (idx >= BATCH * 2 * HSM) return;
  int b = idx / (2 * HSM);
  int r = idx - b * (2 * HSM);
  int m = r / HSM;
  int e = r - m * HSM;
  hbuf[(size_t)b * HSB + m * HSM + e] = (m == 0) ? 0.f : img[(size_t)b * HSM + e];
}

// ---------------------------------------------------------------------------
// One full module step per workgroup (grid = B = 128, block = 256 = 8 waves).
// ---------------------------------------------------------------------------
__global__ __launch_bounds__(256) void shnmn_module_kernel(
    const float* __restrict__ question,
    const float* __restrict__ alpha,
    const float* __restrict__ tau0,
    const float* __restrict__ tau1,
    float* __restrict__ hbuf,
    int mod)
{
  extern __shared__ float smem[];
  float* lds_x   = smem + OFF_X;
  float* lds_pw  = smem + OFF_PW;
  float* lds_w   = smem + OFF_W;
  float* lds_inp = smem + OFF_INP;
  float* lds_pb  = smem + OFF_PB;
  float* lds_cb  = smem + OFF_CB;

  const int b    = blockIdx.x;
  const int tid  = threadIdx.x;
  const int lane = tid & 31;
  const int wave = tid >> 5;
  const int ln   = lane & 15;          // N index (B/C) and M index (A) in tile
  const int lh   = lane >> 4;          // lane-half: K group (A/B), M+8 (C)
  const int nmix = mod + 2;

  float aw[3];  softmaxn(alpha + mod * 3, 3, aw);
  float t0w[4]; softmaxn(tau0 + mod * (NMOD + 1), nmix, t0w);
  float t1w[4]; softmaxn(tau1 + mod * (NMOD + 1), nmix, t1w);

  const float*  qb = question + (size_t)b * (NMOD * QDIM);
  const float4* q4 = (const float4*)qb;
  const float*  hb = hbuf + (size_t)b * HSB;
  const float4* h4 = (const float4*)hb;

  // =================== phase 0: stage x, proj_w, biases ====================
  // Warm L2/WGP$ with the conv-weight region consumed in phase 2.
  for (int e = tid * 64; e < CW; e += 256 * 64) {
    __builtin_prefetch(qb + e, 0, 0);
    __builtin_prefetch(qb + QDIM + e, 0, 0);
    __builtin_prefetch(qb + 2 * QDIM + e, 0, 0);
  }

  // x = [tau0-mix ; tau1-mix], pair-interleaved over channel: [(ch>>1)][p][ch&1]
  for (int e4 = tid; e4 < 128 * (NPIX / 4); e4 += 256) {
    const int ch = e4 >> 6;
    const int p  = (e4 & 63) * 4;
    const float* tw = (ch < DIMC) ? t0w : t1w;
    const int c = ch & 63;
    float4 s = {0.f, 0.f, 0.f, 0.f};
    for (int m = 0; m < nmix; ++m) {
      const float4 hv = h4[(m * HSM + c * NPIX + p) >> 2];
      const float w = tw[m];
      s.x += w * hv.x; s.y += w * hv.y; s.z += w * hv.z; s.w += w * hv.w;
    }
    const int base = (ch >> 1) * (2 * NPIX) + p * 2 + (ch & 1);
    lds_x[base]     = s.x;
    lds_x[base + 2] = s.y;
    lds_x[base + 4] = s.z;
    lds_x[base + 6] = s.w;
  }

  // proj_w mixed + transposed, pair-interleaved over K=i2: [(i2>>1)][o][i2&1]
  for (int e4 = tid; e4 < (2 * DIMC * DIMC) / 4; e4 += 256) {
    const float4 w0 = q4[(PWOFF >> 2) + e4];
    const float4 w1 = q4[((QDIM + PWOFF) >> 2) + e4];
    const float4 w2 = q4[((2 * QDIM + PWOFF) >> 2) + e4];
    const int e  = e4 * 4;
    const int o  = e >> 7;             // 128 i2 per o, group of 4 never crosses o
    const int i2 = e & 127;            // even
    const int base = (i2 >> 1) * 128 + o * 2;
    lds_pw[base]       = aw[0] * w0.x + aw[1] * w1.x + aw[2] * w2.x;
    lds_pw[base + 1]   = aw[0] * w0.y + aw[1] * w1.y + aw[2] * w2.y;
    lds_pw[base + 128] = aw[0] * w0.z + aw[1] * w1.z + aw[2] * w2.z;
    lds_pw[base + 129] = aw[0] * w0.w + aw[1] * w1.w + aw[2] * w2.w;
  }

  if (tid < DIMC) {
    lds_pb[tid] = aw[0] * qb[PBOFF + tid] + aw[1] * qb[QDIM + PBOFF + tid]
                + aw[2] * qb[2 * QDIM + PBOFF + tid];
    lds_cb[tid] = aw[0] * qb[CW + tid] + aw[1] * qb[QDIM + CW + tid]
                + aw[2] * qb[2 * QDIM + CW + tid];
  }

  // zero padded inp planes (borders stay zero; interior overwritten later)
  for (int e = tid; e < 32 * (2 * PADP); e += 256) lds_inp[e] = 0.f;

  __syncthreads();

  // ============ phase 1: proj GEMM  M=64, K=128, N=256 (WMMA f32) ==========
  const int mt  = wave & 3;            // M tile (out channels mt*16..+15)
  const int nt0 = (wave >> 2) * 8;     // 8 N tiles (rows) per wave
  const int o   = mt * 16 + ln;        // A-frag m = lane&15

  v8f acc[8];
  for (int q = 0; q < 8; ++q) acc[q] = v8f{};

  for (int kb = 0; kb < 128; kb += 4) {
    const int kp = (kb >> 1) + lh;     // pair index of k0 = kb + 2*lh
    const v2f a = ldsp(lds_pw + kp * 128 + o * 2);
#pragma unroll
    for (int q = 0; q < 8; ++q) {
      const int p = (nt0 + q) * 16 + ln;
      const v2f bf = ldsp(lds_x + kp * (2 * NPIX) + p * 2);
      acc[q] = __builtin_amdgcn_wmma_f32_16x16x4_f32(
          false, a, false, bf, (short)0, acc[q], false, false);
    }
  }

  // epilogue 1: inp = acc + proj_b  ->  padded LDS, ds_store_b64 channel pairs
#pragma unroll
  for (int q = 0; q < 8; ++q) {
    const int y    = nt0 + q;
    const int pix2 = ((y + 1) * 18 + (ln + 1)) * 2;
#pragma unroll
    for (int v = 0; v < 8; v += 2) {
      const int oc = mt * 16 + 8 * lh + v;     // even: C layout M = vgpr + 8*half
      v2f val;
      val.x = acc[q][v]     + lds_pb[oc];
      val.y = acc[q][v + 1] + lds_pb[oc + 1];
      *(v2f*)(lds_inp + (oc >> 1) * (2 * PADP) + pix2) = val;
    }
  }

  __syncthreads();   // all inp written; x/pw region now dead

  // ============ phase 2a: stage mixed cnn_w over the dead region ===========
  // layout: [(c>>1)][t][o][c&1]  -> A-frag pair is one ds_load_b64
  for (int e4 = tid; e4 < CW / 4; e4 += 256) {
    const float4 w0 = q4[e4];
    const float4 w1 = q4[(QDIM >> 2) + e4];
    const float4 w2 = q4[(2 * QDIM >> 2) + e4];
    const int e   = e4 * 4;
    const int oo  = e / 576;           // 576 floats per o; groups of 4 never cross
    const int rem = e - oo * 576;      // rem = c*9 + t
    const float s[4] = {
        aw[0] * w0.x + aw[1] * w1.x + aw[2] * w2.x,
        aw[0] * w0.y + aw[1] * w1.y + aw[2] * w2.y,
        aw[0] * w0.z + aw[1] * w1.z + aw[2] * w2.z,
        aw[0] * w0.w + aw[1] * w1.w + aw[2] * w2.w};
#pragma unroll
    for (int j = 0; j < 4; ++j) {
      const int r = rem + j;
      const int c = r / 9;
      const int t = r - c * 9;
      lds_w[((c >> 1) * 9 + t) * 128 + oo * 2 + (c & 1)] = s[j];
    }
  }

  __syncthreads();

  // ====== phase 2b: conv as implicit GEMM  M=64, K=576 (t outer), N=256 ====
  for (int q = 0; q < 8; ++q) acc[q] = v8f{};

  for (int t = 0; t < 9; ++t) {
    const int dy  = t / 3 - 1;
    const int dx  = t - (t / 3) * 3 - 1;
    const int col = ln + dx + 1;
    for (int cb = 0; cb < 64; cb += 4) {
      const int cp = (cb >> 1) + lh;   // pair index of c0 = cb + 2*lh
      const v2f a = ldsp(lds_w + (cp * 9 + t) * 128 + o * 2);
#pragma unroll
      for (int q = 0; q < 8; ++q) {
        const int row = nt0 + q + dy + 1;
        const v2f bf = ldsp(lds_inp + cp * (2 * PADP) + (row * 18 + col) * 2);
        acc[q] = __builtin_amdgcn_wmma_f32_16x16x4_f32(
            false, a, false, bf, (short)0, acc[q], false, false);
      }
    }
  }

  // epilogue 2: +cnn_b, ReLU, write h slot mod+2 (coalesced 16-float rows)
  float* ob = hbuf + (size_t)b * HSB + (mod + 2) * HSM;
#pragma unroll
  for (int q = 0; q < 8; ++q) {
    const int y = nt0 + q;
#pragma unroll
    for (int v = 0; v < 8; ++v) {
      const int oc = mt * 16 + 8 * lh + v;
      ob[oc * NPIX + y * 16 + ln] = fmaxf(acc[q][v] + lds_cb[oc], 0.f);
    }
  }
}

// ---------------------------------------------------------------------------
extern "C" void kernel_launch(void* const* d_in, const int* in_sizes, int n_in,
                              void* d_out, int out_size, void* d_ws, size_t ws_size,
                              hipStream_t stream) {
  const float* question = (const float*)d_in[0];   // (B, 3, QDIM)
  const float* img      = (const float*)d_in[1];   // (B, 64, 16, 16)
  const float* alpha    = (const float*)d_in[2];   // (3, 3)
  const float* tau0     = (const float*)d_in[3];   // (3, 4)
  const float* tau1     = (const float*)d_in[4];   // (3, 4)

  float* hbuf = (float*)d_out;    // (B, 1, 5, 64, 16, 16) doubles as h_prev

  {
    const int total = BATCH * 2 * HSM;
    init_h_kernel<<<(total + 255) / 256, 256, 0, stream>>>(img, hbuf);
  }

  const size_t shm = (size_t)LDS_FLOATS * sizeof(float);   // ~241.5 KB
  for (int mod = 0; mod < NMOD; ++mod) {
    shnmn_module_kernel<<<BATCH, 256, shm, stream>>>(question, alpha, tau0,
                                                     tau1, hbuf, mod);
  }
}